// Attention_61237643706555
// MI455X (gfx1250) — compile-verified
//
#include <hip/hip_runtime.h>
#include <math.h>

#define BS 8
#define NE 96
#define Dm 512
#define NH 8
#define DK 64
#define ITILES (NE/16)              // 6 i-tiles of 16
#define NROWS (BS*NE*NH)            // 6144 e-chunk rows, layout [b][i][h][c]
#define NROWS2 (BS*NH*NE)           // 6144 T rows, layout [b][h][j][c]

typedef __attribute__((ext_vector_type(2))) float v2f;
typedef __attribute__((ext_vector_type(8))) float v8f;

// ---- workspace layout (in floats) ----
#define OFF_G   0u            // g  [NROWS][64]
#define OFF_F   393216u       // f  [NROWS][64]
#define OFF_EV  786432u       // ev [NROWS][64]
#define OFF_TP  1179648u      // Tpart [ITILES][NROWS2][64]
#define OFF_SP  3538944u      // spart [ITILES][NROWS2]
#define OFF_CA  3575808u      // calpha [NROWS]
#define OFF_CB  3581952u      // cbeta  [NROWS]
#define OFF_MA  3588096u      // W_K^T W_Q  [64][64]
#define OFF_MB  3592192u      // W_Q^T W_K  [64][64]
#define OFF_U1  3596288u      // W_K^T b_Q  [64]
#define OFF_U2  3596352u      // W_Q^T b_K  [64]
#define OFF_Q   3596416u      // b_Q . b_K

// ============================================================ kernel A0
__global__ __launch_bounds__(256) void prep_mats(
    const float* __restrict__ WQw, const float* __restrict__ WQb,
    const float* __restrict__ WKw, const float* __restrict__ WKb,
    float* __restrict__ MA, float* __restrict__ MB,
    float* __restrict__ u1, float* __restrict__ u2, float* __restrict__ q)
{
    const int t = threadIdx.x;
    for (int idx = t; idx < 64*64; idx += 256) {
        int c = idx >> 6, k = idx & 63;
        float sA = 0.f, sB = 0.f;
        for (int tt = 0; tt < 64; ++tt) {
            sA += WKw[tt*64 + c] * WQw[tt*64 + k];   // (W_K^T W_Q)[c,k]
            sB += WQw[tt*64 + c] * WKw[tt*64 + k];   // (W_Q^T W_K)[c,k]
        }
        MA[idx] = sA; MB[idx] = sB;
    }
    if (t < 64) {
        float s1 = 0.f, s2 = 0.f;
        for (int tt = 0; tt < 64; ++tt) {
            s1 += WKw[tt*64 + t] * WQb[tt];
            s2 += WQw[tt*64 + t] * WKb[tt];
        }
        u1[t] = s1; u2[t] = s2;
    }
    if (t == 0) {
        float s = 0.f;
        for (int tt = 0; tt < 64; ++tt) s += WQb[tt] * WKb[tt];
        q[0] = s;
    }
}

// ============================================================ WMMA helper
// out[r, c] = sum_k mat[c,k] * A[r,k] + bias[c]   (16-row tile, K=N=64)
__device__ __forceinline__ void wmma_proj_one(
    const v2f aF[16], const float* __restrict__ mat,
    const float* __restrict__ bias, float* __restrict__ outp,
    int r0, int lane)
{
    const int  nlow = lane & 15;
    const bool lo   = lane < 16;
#pragma unroll
    for (int nt = 0; nt < 4; ++nt) {
        v8f acc = {0.f,0.f,0.f,0.f,0.f,0.f,0.f,0.f};
        const int n = nt*16 + nlow;
        const float* mrow = mat + n*64;         // B[k][n] = mat[n][k]
#pragma unroll
        for (int ks = 0; ks < 16; ++ks) {
            const int kb = ks*4 + (lo ? 0 : 2);
            v2f bf = *(const v2f*)(mrow + kb);
            acc = __builtin_amdgcn_wmma_f32_16x16x4_f32(
                false, aF[ks], false, bf, (short)0, acc, false, false);
        }
        const int col = nt*16 + nlow;
        const float bc = bias[col];
#pragma unroll
        for (int v = 0; v < 8; ++v) {
            const int row = r0 + v + (lo ? 0 : 8);
            outp[(size_t)row*64 + col] = acc[v] + bc;
        }
    }
}

// ============================================================ kernel A1
// g = (W_K^T W_Q) e + u1 ; f = (W_Q^T W_K) e + u2 ; ev = W_V e + b_V
__global__ __launch_bounds__(128) void proj_wmma(
    const float* __restrict__ E,
    const float* __restrict__ MA, const float* __restrict__ MB,
    const float* __restrict__ WVw,
    const float* __restrict__ u1, const float* __restrict__ u2,
    const float* __restrict__ WVb,
    float* __restrict__ g, float* __restrict__ f, float* __restrict__ ev)
{
    const int lane = threadIdx.x & 31;
    const int wave = threadIdx.x >> 5;
    const int r0   = (blockIdx.x*4 + wave)*16;
    const int arow = r0 + (lane & 15);
    const float* erow = E + (size_t)arow*64;
    v2f aF[16];
#pragma unroll
    for (int ks = 0; ks < 16; ++ks) {
        const int kb = ks*4 + ((lane < 16) ? 0 : 2);
        aF[ks] = *(const v2f*)(erow + kb);
    }
    wmma_proj_one(aF, MA,  u1,  g,  r0, lane);
    wmma_proj_one(aF, MB,  u2,  f,  r0, lane);
    wmma_proj_one(aF, WVw, WVb, ev, r0, lane);
}

// ============================================================ kernel A2
__global__ __launch_bounds__(256) void consts_k(
    const float* __restrict__ E,
    const float* __restrict__ u1, const float* __restrict__ u2,
    const float* __restrict__ q,
    float* __restrict__ calpha, float* __restrict__ cbeta)
{
    const int r = blockIdx.x*256 + threadIdx.x;
    if (r >= NROWS) return;
    const float* e = E + (size_t)r*64;
    float d2 = 0.f, d1 = 0.f;
#pragma unroll 8
    for (int c = 0; c < 64; ++c) { float x = e[c]; d2 += x*u2[c]; d1 += x*u1[c]; }
    const float qq = q[0];
    calpha[r] = d2 + qq;
    cbeta[r]  = d1 + qq;
}

// ============================================================ kernel B (bandwidth kernel)
__global__ __launch_bounds__(256) void rel_main(
    const float* __restrict__ R, const float* __restrict__ Rmas,
    const float* __restrict__ Emas,
    const float* __restrict__ g, const float* __restrict__ f,
    const float* __restrict__ ev,
    const float* __restrict__ calpha, const float* __restrict__ cbeta,
    float* __restrict__ Tpart, float* __restrict__ spart,
    float* __restrict__ RZ)
{
    __shared__ float sh_f[NE][DK];     // f[j] for all j of this (b,h)
    __shared__ float sh_g[16][DK];     // g[i] for this i-tile
    __shared__ float sh_a[16][NE];     // logits -> normalized alpha
    __shared__ float sh_b0[16][NE];    // beta softmax weight b0
    __shared__ float sh_cb[NE];
    __shared__ float sh_ca[16];

    const int t   = threadIdx.x;
    const int bid = blockIdx.x;
    const int it  = bid % ITILES;
    const int h   = (bid / ITILES) % NH;
    const int b   = bid / (ITILES*NH);
    const int i0  = it*16;

    for (int idx = t; idx < NE*DK; idx += 256) {
        int j = idx >> 6, c = idx & 63;
        sh_f[j][c] = f[(((size_t)b*NE + j)*NH + h)*DK + c];
    }
    for (int idx = t; idx < 16*DK; idx += 256) {
        int ii = idx >> 6, c = idx & 63;
        sh_g[ii][c] = g[(((size_t)b*NE + i0 + ii)*NH + h)*DK + c];
    }
    if (t < NE) sh_cb[t] = cbeta[((size_t)b*NE + t)*NH + h];
    if (t < 16) sh_ca[t] = calpha[((size_t)b*NE + i0 + t)*NH + h];
    __syncthreads();

    // ---- pass 1: alpha logits + pairwise beta softmax, fused over one read of X
    for (int r = 0; r < 6; ++r) {
        const int task = r*256 + t;            // 16*96 tasks
        const int ii = task / NE, j = task % NE;
        const float4* X = (const float4*)(R + ((((size_t)b*NE + (i0+ii))*NE + j)*Dm) + h*DK);
        __builtin_prefetch(X + 16, 0, 0);
        float dg = 0.f, d0 = 0.f, d1 = 0.f;
#pragma unroll
        for (int c4 = 0; c4 < 16; ++c4) {
            const float4 x = X[c4];
            const int c = c4*4;
            dg += x.x*sh_g[ii][c] + x.y*sh_g[ii][c+1] + x.z*sh_g[ii][c+2] + x.w*sh_g[ii][c+3];
            d0 += x.x*sh_f[i0+ii][c] + x.y*sh_f[i0+ii][c+1] + x.z*sh_f[i0+ii][c+2] + x.w*sh_f[i0+ii][c+3];
            d1 += x.x*sh_f[j][c] + x.y*sh_f[j][c+1] + x.z*sh_f[j][c+2] + x.w*sh_f[j][c+3];
        }
        float lg = dg + sh_ca[ii];
        if (!(Emas[b*NE + j] > 0.f)) lg = -__builtin_inff();
        sh_a[ii][j] = lg;
        const float b0v = d0 + sh_cb[i0+ii];
        const float b1v = d1 + sh_cb[j];
        const float mm = fmaxf(b0v, b1v);
        const float e0 = __expf(b0v - mm), e1 = __expf(b1v - mm);
        sh_b0[ii][j] = e0 / (e0 + e1);
    }
    __syncthreads();

    // ---- pass 1.5: softmax over j, per row
    if (t < 16) {
        float m = -__builtin_inff();
        for (int j = 0; j < NE; ++j) m = fmaxf(m, sh_a[t][j]);
        float s = 0.f;
        if (m > -__builtin_inff())
            for (int j = 0; j < NE; ++j) s += __expf(sh_a[t][j] - m);
        const float rinv = (s > 0.f) ? 1.f/s : 0.f;
        for (int j = 0; j < NE; ++j) {
            float v = (m > -__builtin_inff()) ? __expf(sh_a[t][j] - m)*rinv : 0.f;
            sh_a[t][j] = v;
        }
    }
    __syncthreads();

    // ---- pass 2: per (j, 16B group) accumulate T over i, emit R_Z
    const int cg = t & 15;
    const int jr = t >> 4;
    for (int jb = 0; jb < 6; ++jb) {
        const int j = jb*16 + jr;
        const float4 evj = *(const float4*)(ev + (((size_t)b*NE + j)*NH + h)*DK + cg*4);
        float4 acc = make_float4(0.f, 0.f, 0.f, 0.f);
        float asum = 0.f;
        for (int ii = 0; ii < 16; ++ii) {
            const int i = i0 + ii;
            const size_t choff = ((((size_t)b*NE + i)*NE + j)*Dm) + h*DK + cg*4;
            if (ii + 1 < 16)
                __builtin_prefetch(R + choff + (size_t)NE*Dm, 0, 0);
            const float4 x = *(const float4*)(R + choff);
            const float a = sh_a[ii][j];
            acc.x += a*x.x; acc.y += a*x.y; acc.z += a*x.z; acc.w += a*x.w;
            asum += a;
            const float b0v = sh_b0[ii][j], b1v = 1.f - b0v;
            const float4 evi = *(const float4*)(ev + (((size_t)b*NE + i)*NH + h)*DK + cg*4);
            float4 rz;
            rz.x = b0v*evi.x + b1v*evj.x;
            rz.y = b0v*evi.y + b1v*evj.y;
            rz.z = b0v*evi.z + b1v*evj.z;
            rz.w = b0v*evi.w + b1v*evj.w;
            if (!(Rmas[((size_t)b*NE + i)*NE + j] > 0.f)) {
                rz.x = 0.f; rz.y = 0.f; rz.z = 0.f; rz.w = 0.f;
            }
            *(float4*)(RZ + choff) = rz;
        }
        const size_t sidx = ((size_t)b*NH + h)*NE + j;
        *(float4*)(Tpart + ((size_t)it*NROWS2 + sidx)*DK + cg*4) = acc;
        if (cg == 0) spart[(size_t)it*NROWS2 + sidx] = asum;
    }
}

// ============================================================ kernel C
// E_Z[j,c] = sum_k WV[c,k]*T[j,k] + s[j]*bV[c], masked
__global__ __launch_bounds__(128) void ez_wmma(
    const float* __restrict__ Tpart, const float* __restrict__ spart,
    const float* __restrict__ WVw, const float* __restrict__ WVb,
    const float* __restrict__ Emas, float* __restrict__ EZ)
{
    const int lane = threadIdx.x & 31;
    const int wave = threadIdx.x >> 5;
    const int r0   = (blockIdx.x*4 + wave)*16;
    const int arow = r0 + (lane & 15);
    const bool lo  = lane < 16;
    v2f aF[16];
#pragma unroll
    for (int ks = 0; ks < 16; ++ks) {
        const int kb = ks*4 + (lo ? 0 : 2);
        float a0 = 0.f, a1 = 0.f;
#pragma unroll
        for (int p = 0; p < ITILES; ++p) {
            const float* tp = Tpart + ((size_t)p*NROWS2 + arow)*DK + kb;
            a0 += tp[0]; a1 += tp[1];
        }
        aF[ks].x = a0; aF[ks].y = a1;
    }
    const int nlow = lane & 15;
#pragma unroll
    for (int nt = 0; nt < 4; ++nt) {
        v8f acc = {0.f,0.f,0.f,0.f,0.f,0.f,0.f,0.f};
        const int n = nt*16 + nlow;
        const float* mrow = WVw + n*64;
#pragma unroll
        for (int ks = 0; ks < 16; ++ks) {
            const int kb = ks*4 + (lo ? 0 : 2);
            v2f bf = *(const v2f*)(mrow + kb);
            acc = __builtin_amdgcn_wmma_f32_16x16x4_f32(
                false, aF[ks], false, bf, (short)0, acc, false, false);
        }
        const int col = nt*16 + nlow;
        const float bc = WVb[col];
#pragma unroll
        for (int v = 0; v < 8; ++v) {
            const int row = r0 + v + (lo ? 0 : 8);
            float sv = 0.f;
#pragma unroll
            for (int p = 0; p < ITILES; ++p) sv += spart[(size_t)p*NROWS2 + row];
            const int bb = row / (NH*NE);
            const int hh = (row / NE) % NH;
            const int jj = row % NE;
            float val = acc[v] + sv*bc;
            if (!(Emas[bb*NE + jj] > 0.f)) val = 0.f;
            EZ[((size_t)bb*NE + jj)*Dm + hh*DK + col] = val;
        }
    }
}

// ============================================================ launch
extern "C" void kernel_launch(void* const* d_in, const int* in_sizes, int n_in,
                              void* d_out, int out_size, void* d_ws, size_t ws_size,
                              hipStream_t stream) {
    (void)in_sizes; (void)n_in; (void)out_size; (void)ws_size;
    const float* R    = (const float*)d_in[0];
    const float* E    = (const float*)d_in[1];
    const float* Rmas = (const float*)d_in[2];
    const float* Emas = (const float*)d_in[3];
    const float* WQw  = (const float*)d_in[4];
    const float* WQb  = (const float*)d_in[5];
    const float* WKw  = (const float*)d_in[6];
    const float* WKb  = (const float*)d_in[7];
    const float* WVw  = (const float*)d_in[8];
    const float* WVb  = (const float*)d_in[9];

    float* ws = (float*)d_ws;
    float* RZ = (float*)d_out;
    float* EZ = (float*)d_out + (size_t)BS*NE*NE*Dm;

    prep_mats<<<1, 256, 0, stream>>>(WQw, WQb, WKw, WKb,
        ws+OFF_MA, ws+OFF_MB, ws+OFF_U1, ws+OFF_U2, ws+OFF_Q);

    proj_wmma<<<NROWS/16/4, 128, 0, stream>>>(E,
        ws+OFF_MA, ws+OFF_MB, WVw, ws+OFF_U1, ws+OFF_U2, WVb,
        ws+OFF_G, ws+OFF_F, ws+OFF_EV);

    consts_k<<<(NROWS + 255)/256, 256, 0, stream>>>(E,
        ws+OFF_U1, ws+OFF_U2, ws+OFF_Q, ws+OFF_CA, ws+OFF_CB);

    rel_main<<<BS*NH*ITILES, 256, 0, stream>>>(R, Rmas, Emas,
        ws+OFF_G, ws+OFF_F, ws+OFF_EV, ws+OFF_CA, ws+OFF_CB,
        ws+OFF_TP, ws+OFF_SP, RZ);

    ez_wmma<<<NROWS2/16/4, 128, 0, stream>>>(ws+OFF_TP, ws+OFF_SP,
        WVw, WVb, Emas, EZ);
}